// FullAttention_v5_47330539602003
// MI455X (gfx1250) — compile-verified
//
#include <hip/hip_runtime.h>
#include <hip/hip_bf16.h>

typedef __attribute__((ext_vector_type(2))) float v2f;
typedef __attribute__((ext_vector_type(4))) float v4f;
typedef __attribute__((ext_vector_type(8))) float v8f;

namespace {
constexpr int kB = 4480;
constexpr int kL = 7;
constexpr int kH = 8;
constexpr int kE = 64;
constexpr int kS = 56;
constexpr int kD = 64;
constexpr int kNSt = 35;
constexpr float kScale = 0.125f;      // 1/sqrt(64)
constexpr float kNoiseStd = 0.1f;
}

// One wave32 per (b, h). fp32 end-to-end via V_WMMA_F32_16X16X4_F32.
__global__ __launch_bounds__(32)
void fullattn_wmma_f32_kernel(const float* __restrict__ q,
                              const float* __restrict__ kmat,
                              const float* __restrict__ vmat,
                              const float* __restrict__ pmat,
                              const float* __restrict__ zmat,
                              float* __restrict__ out)
{
    // Row pitches chosen for conflict-free ds_load_b64 fragment gathers.
    __shared__ float Qs[16][68];   // Q rows 0..6 valid, 7..15 zero
    __shared__ float Ks[64][68];   // K rows 0..55 valid, 56..63 zero
    __shared__ float A1[16][60];   // softmax * clamp(P)           (rows 7..15 zero)
    __shared__ float A2[16][60];   // softmax * (1-clamp(P)) * 0.1 (rows 7..15 zero)

    const int bh   = blockIdx.x;
    const int b    = bh / kH;
    const int h    = bh % kH;
    const int n    = b % kNSt;             // station id
    const int lane = threadIdx.x;
    const int g    = lane >> 4;            // lane group 0/1
    const int m    = lane & 15;            // row-in-group

    // ---- Stage Q[7][64] (coalesced float4), zero pad rows 7..15 ----
    for (int i = lane; i < kL * 16; i += 32) {
        const int r = i >> 4, c = (i & 15) << 2;
        const v4f val = *(const v4f*)(q + (((size_t)b * kL + r) * kH + h) * kE + c);
        Qs[r][c + 0] = val.x; Qs[r][c + 1] = val.y;
        Qs[r][c + 2] = val.z; Qs[r][c + 3] = val.w;
    }
    for (int i = lane; i < 9 * 16; i += 32) {
        const int r = 7 + (i >> 4), c = (i & 15) << 2;
        Qs[r][c + 0] = 0.f; Qs[r][c + 1] = 0.f; Qs[r][c + 2] = 0.f; Qs[r][c + 3] = 0.f;
    }
    // ---- Stage K[56][64] (coalesced float4), zero pad rows 56..63 ----
    for (int i = lane; i < kS * 16; i += 32) {
        const int r = i >> 4, c = (i & 15) << 2;
        const v4f val = *(const v4f*)(kmat + (((size_t)b * kS + r) * kH + h) * kE + c);
        Ks[r][c + 0] = val.x; Ks[r][c + 1] = val.y;
        Ks[r][c + 2] = val.z; Ks[r][c + 3] = val.w;
    }
    for (int i = lane; i < 8 * 68; i += 32)
        Ks[56 + i / 68][i % 68] = 0.f;
    // ---- Zero pad rows 7..15 of A1/A2 (pad M-rows feed the output GEMM) ----
    for (int i = lane; i < 9 * 60; i += 32) {
        A1[7 + i / 60][i % 60] = 0.f;
        A2[7 + i / 60][i % 60] = 0.f;
    }
    __syncthreads();

    // ================= Phase 1: scores = Q @ K^T  (16x16x4 f32 WMMA) =======
    // A frag: lane -> M=m, vgpr j -> K = e0 + 2g + j   (contiguous pair)
    // B frag: lane -> N=m, vgpr j -> K = e0 + 2g + j ; B[e][s] = K[s][e]
    v8f accS[4];
    accS[0] = accS[1] = accS[2] = accS[3] = (v8f)0.f;
#pragma unroll
    for (int e0 = 0; e0 < kE; e0 += 4) {
        v2f a;
        a.x = Qs[m][e0 + 2 * g];
        a.y = Qs[m][e0 + 2 * g + 1];
#pragma unroll
        for (int t = 0; t < 4; ++t) {
            v2f bf;
            bf.x = Ks[t * 16 + m][e0 + 2 * g];
            bf.y = Ks[t * 16 + m][e0 + 2 * g + 1];
            accS[t] = __builtin_amdgcn_wmma_f32_16x16x4_f32(
                false, a, false, bf, (short)0, accS[t], false, false);
        }
    }

    // ---- Spill scaled scores into A1 (C/D layout: lane->(N=s), vgpr v -> M=v+8g)
#pragma unroll
    for (int t = 0; t < 4; ++t) {
        const int s = t * 16 + m;
        if (s < kS) {
#pragma unroll
            for (int vv = 0; vv < 8; ++vv)
                A1[vv + 8 * g][s] = accS[t][vv] * kScale;
        }
    }
    __syncthreads();

    // ================= Phase 2: row softmax + blend with prob matrix =======
    if (lane < kL) {
        const float* prow = pmat + ((size_t)n * kL + lane) * kS;
        float mx = -3.4e38f;
        for (int s = 0; s < kS; ++s) mx = fmaxf(mx, A1[lane][s]);
        float sum = 0.f;
        for (int s = 0; s < kS; ++s) {
            const float e = __expf(A1[lane][s] - mx);
            A1[lane][s] = e;
            sum += e;
        }
        const float rs = 1.f / sum;
        for (int s = 0; s < kS; ++s) {
            const float a = A1[lane][s] * rs;
            const float p = fminf(fmaxf(prow[s], 0.f), 1.f);
            A1[lane][s] = a * p;
            A2[lane][s] = a * (1.f - p) * kNoiseStd;   // fold 0.1*noise scale here
        }
    }
    __syncthreads();

    // ================= Phase 3: out = A1 @ V + A2 @ Z ======================
    // V/Z B-fragments read straight from global: lanes 0..15 -> contiguous d,
    // lanes 16..31 -> contiguous d at s+2: two 64B segments per load. Each
    // element of the 514MB V / noise arrays is touched exactly once.
    const size_t vzoff = ((size_t)b * kS * kH + h) * kD;   // + s*kH*kD + d
    const float* Vb = vmat + vzoff;
    const float* Zb = zmat + vzoff;

    v8f accO[4];
    accO[0] = accO[1] = accO[2] = accO[3] = (v8f)0.f;
#pragma unroll
    for (int s0 = 0; s0 < kS; s0 += 4) {
        const int kk = s0 + 2 * g;           // contraction rows kk, kk+1 (<=55)
        v2f a1, a2;
        a1.x = A1[m][kk]; a1.y = A1[m][kk + 1];
        a2.x = A2[m][kk]; a2.y = A2[m][kk + 1];
#pragma unroll
        for (int t = 0; t < 4; ++t) {
            const int d = t * 16 + m;
            v2f bv, bz;
            bv.x = Vb[(size_t)kk * (kH * kD) + d];
            bv.y = Vb[(size_t)(kk + 1) * (kH * kD) + d];
            bz.x = Zb[(size_t)kk * (kH * kD) + d];
            bz.y = Zb[(size_t)(kk + 1) * (kH * kD) + d];
            accO[t] = __builtin_amdgcn_wmma_f32_16x16x4_f32(
                false, a1, false, bv, (short)0, accO[t], false, false);
            accO[t] = __builtin_amdgcn_wmma_f32_16x16x4_f32(
                false, a2, false, bz, (short)0, accO[t], false, false);
        }
    }

    // ---- Store out[b,l,h,d]: rows l = vv + 8g < 7, lanes give contiguous d
#pragma unroll
    for (int t = 0; t < 4; ++t) {
        const int d = t * 16 + m;
#pragma unroll
        for (int vv = 0; vv < 8; ++vv) {
            const int l = vv + 8 * g;
            if (l < kL)
                out[(((size_t)b * kL + l) * kH + h) * kD + d] = accO[t][vv];
        }
    }
}

extern "C" void kernel_launch(void* const* d_in, const int* in_sizes, int n_in,
                              void* d_out, int out_size, void* d_ws, size_t ws_size,
                              hipStream_t stream) {
    const float* q    = (const float*)d_in[0];   // [B, L, H, E]
    const float* kmat = (const float*)d_in[1];   // [B, S, H, E]
    const float* vmat = (const float*)d_in[2];   // [B, S, H, D]
    const float* pmat = (const float*)d_in[3];   // [N, L, S]
    const float* zmat = (const float*)d_in[4];   // [AB, N, S, H, D] == flat [B, S, H, D]
    float* out = (float*)d_out;

    dim3 grid(kB * kH);   // one wave32 per (b, h)
    fullattn_wmma_f32_kernel<<<grid, 32, 0, stream>>>(q, kmat, vmat, pmat, zmat, out);
}